// MarginRankingLoss_8796093022297
// MI455X (gfx1250) — compile-verified
//
#include <hip/hip_runtime.h>

// CDNA5 wave32 WMMA types
typedef __attribute__((ext_vector_type(2))) float v2f;  // 16x4 / 4x16 f32 operand: 2 VGPRs
typedef __attribute__((ext_vector_type(8))) float v8f;  // 16x16 f32 C/D: 8 VGPRs

static constexpr int   kBlock        = 256;             // 8 waves per block (wave32)
static constexpr int   kWavesPerBlk  = kBlock / 32;
static constexpr int   kRowsPerWave  = 16;
static constexpr int   kRowsPerBlk   = kWavesPerBlk * kRowsPerWave; // 128
static constexpr int   kC            = 64;
static constexpr float kMargin       = 0.01f;
static constexpr float kBigF         = 3.0e38f;         // > any |diff|, < inf

// Single-instruction relu (v_med3) for the scalar summary term.
__device__ __forceinline__ float relu1(float x) {
  return __builtin_amdgcn_fmed3f(x, 0.0f, kBigF);
}

// acc += relu(d), rewritten as acc = max(acc + d, acc).
// Exact in f32: d>0 -> both forms give fl(acc+d); d<=0 -> fl(acc+d) <= acc
// under RNE so max returns acc, identical to adding relu(d)=0.
// __builtin_fmaxf lowers inline to llvm.maxnum, so the fast-math pragma's
// nnan flag attaches here (unlike the OCML fmaxf call) and the backend can
// drop the canonicalize it otherwise inserts for the loop-carried acc PHI.
// All values are finite by construction.
#pragma float_control(push)
#pragma float_control(precise, off)
__device__ __forceinline__ void relu_acc(v8f& acc, v8f d) {
  const v8f t = acc + d;            // v_pk_add_f32 x4
#pragma unroll
  for (int k = 0; k < 8; ++k)
    acc[k] = __builtin_fmaxf(t[k], acc[k]);  // v_max_num_f32 x8 (dual-paired)
}
#pragma float_control(pop)

__device__ __forceinline__ v8f wmma_diff(v2f a, v2f b) {
  const v8f cz = {};
  // D[m,n] = 1*y[n] + (-y[m])*1 = y[n] - y[m]   (exact in f32)
  return __builtin_amdgcn_wmma_f32_16x16x4_f32(false, a, false, b,
                                               (short)0, cz, false, false);
}

__global__ __launch_bounds__(kBlock)
void mrl_pairwise_wmma(const float* __restrict__ cand,
                       const float* __restrict__ summ,
                       float* __restrict__ block_partials,
                       int B) {
  const int  lane  = threadIdx.x & 31;
  const int  wave  = threadIdx.x >> 5;
  const int  lmod  = lane & 15;
  const bool lo    = lane < 16;
  const int  row0  = (blockIdx.x * kWavesPerBlk + wave) * kRowsPerWave;

  // A-matrix VGPR0 is constant across all tiles: K=0 row of ones (lanes 0-15),
  // K=2 row of zeros (lanes 16-31).
  const float a0    = lo ? 1.0f : 0.0f;
  const float mlane = kMargin * (float)lmod;

  // Per-diagonal-offset relu accumulators (t = nb - mb = 0..3).
  v8f acc0 = {}, acc1 = {}, acc2 = {}, acc3 = {};
  float srel = 0.0f;

  for (int r = 0; r < kRowsPerWave; ++r) {
    const int row = row0 + r;          // wave-uniform
    if (row >= B) break;               // uniform branch, EXEC stays all-ones
    const float* __restrict__ xr = cand + (size_t)row * kC;

    // y[c] = x[c] + 0.01*c, fetched as four 16-wide chunks (lanes 16-31 dup).
    const float y0 = xr[ 0 + lmod] + mlane;
    const float y1 = xr[16 + lmod] + (mlane + kMargin * 16.0f);
    const float y2 = xr[32 + lmod] + (mlane + kMargin * 32.0f);
    const float y3 = xr[48 + lmod] + (mlane + kMargin * 48.0f);

    // A operand per m-chunk: {K0=1|K2=0, K1=-y|K3=0}
    const v2f A0 = { a0, lo ? -y0 : 0.0f };
    const v2f A1 = { a0, lo ? -y1 : 0.0f };
    const v2f A2 = { a0, lo ? -y2 : 0.0f };
    const v2f A3 = { a0, lo ? -y3 : 0.0f };
    // B operand per n-chunk: VGPR0 = {K0 row = y[n] (lanes 0-15), K1 row = 1 (lanes 16-31)},
    // VGPR1 = 0 (K2,K3 rows).
    const v2f B0 = { lo ? y0 : 1.0f, 0.0f };
    const v2f B1 = { lo ? y1 : 1.0f, 0.0f };
    const v2f B2 = { lo ? y2 : 1.0f, 0.0f };
    const v2f B3 = { lo ? y3 : 1.0f, 0.0f };

    // Issue all 10 upper-triangular tiles' WMMAs first (independent D regs,
    // back-to-back issue; XDL co-executes with the VALU stream below).
    const v8f d00 = wmma_diff(A0, B0);
    const v8f d11 = wmma_diff(A1, B1);
    const v8f d22 = wmma_diff(A2, B2);
    const v8f d33 = wmma_diff(A3, B3);
    const v8f d01 = wmma_diff(A0, B1);
    const v8f d12 = wmma_diff(A1, B2);
    const v8f d23 = wmma_diff(A2, B3);
    const v8f d02 = wmma_diff(A0, B2);
    const v8f d13 = wmma_diff(A1, B3);
    const v8f d03 = wmma_diff(A0, B3);

    // relu + accumulate (t = nb - mb groups).
    relu_acc(acc0, d00);
    relu_acc(acc0, d11);
    relu_acc(acc0, d22);
    relu_acc(acc0, d33);
    relu_acc(acc1, d01);
    relu_acc(acc1, d12);
    relu_acc(acc1, d23);
    relu_acc(acc2, d02);
    relu_acc(acc2, d13);
    relu_acc(acc3, d03);

    // Summary term: relu(x[c] - s) over all 64 columns (coalesced float2).
    const float  s  = summ[row];
    const float2 xv = *(const float2*)(xr + 2 * lane);
    srel += relu1(xv.x - s) + relu1(xv.y - s);
  }

  // Apply diagonal weights 1/(64-i) once per wave (v_rcp_f32: ~1 ulp, ample).
  // C/D layout: VGPR v, lanes 0-15 -> (M=v, N=lane); lanes 16-31 -> (M=v+8, N=lane-16).
  const int mbase = (lane >> 4) * 8;
  float tot = srel * (1.0f / 64.0f);
#pragma unroll
  for (int v = 0; v < 8; ++v) {
    const int mloc = v + mbase;
#pragma unroll
    for (int t = 0; t < 4; ++t) {
      const int i = 16 * t + lmod - mloc;   // global diagonal offset n - m
      const float a = (t == 0) ? acc0[v] : (t == 1) ? acc1[v]
                    : (t == 2) ? acc2[v] : acc3[v];
      const float w = (i >= 1) ? __builtin_amdgcn_rcpf((float)(64 - i)) : 0.0f;
      tot += a * w;
    }
  }

  // Deterministic reduction: wave shuffle -> LDS -> per-block partial.
#pragma unroll
  for (int off = 16; off > 0; off >>= 1)
    tot += __shfl_xor(tot, off, 32);

  __shared__ float wsum[kWavesPerBlk];
  if (lane == 0) wsum[wave] = tot;
  __syncthreads();
  if (threadIdx.x == 0) {
    float bs = 0.0f;
#pragma unroll
    for (int wv = 0; wv < kWavesPerBlk; ++wv) bs += wsum[wv];
    block_partials[blockIdx.x] = bs;
  }
}

__global__ __launch_bounds__(kBlock)
void mrl_reduce(const float* __restrict__ partials, int n, float inv_b,
                float* __restrict__ out) {
  __shared__ float sm[kBlock];
  float t = 0.0f;
  for (int idx = threadIdx.x; idx < n; idx += kBlock) t += partials[idx];
  sm[threadIdx.x] = t;
  __syncthreads();
  for (int s = kBlock / 2; s > 0; s >>= 1) {
    if (threadIdx.x < s) sm[threadIdx.x] += sm[threadIdx.x + s];
    __syncthreads();
  }
  if (threadIdx.x == 0) out[0] = sm[0] * inv_b;
}

extern "C" void kernel_launch(void* const* d_in, const int* in_sizes, int n_in,
                              void* d_out, int out_size, void* d_ws, size_t ws_size,
                              hipStream_t stream) {
  const float* cand = (const float*)d_in[0];   // (B, 64) f32
  const float* summ = (const float*)d_in[1];   // (B,)   f32
  const int B = in_sizes[1];

  float* partials = (float*)d_ws;              // blocks * 4 bytes of scratch
  const int blocks = (B + kRowsPerBlk - 1) / kRowsPerBlk;   // 1024 for B=131072

  mrl_pairwise_wmma<<<blocks, kBlock, 0, stream>>>(cand, summ, partials, B);
  mrl_reduce<<<1, kBlock, 0, stream>>>(partials, blocks, 1.0f / (float)B,
                                       (float*)d_out);
}